// ExpressionGNN_38208029065790
// MI455X (gfx1250) — compile-verified
//
#include <hip/hip_runtime.h>
#include <math.h>

typedef float v2f __attribute__((ext_vector_type(2)));
typedef float v8f __attribute__((ext_vector_type(8)));

#define D 64
#define NEG_SLOPE 0.2f
#define LN_EPS 1e-5f

// ---------------------------------------------------------------------------
// WMMA GEMM: out[N x 64] = act(A[N x K] @ W[K x 64] + bias)
// One wave computes a 16x16 tile with V_WMMA_F32_16X16X4_F32, K-loop step 4.
// Block = 128 threads (4 waves) = 16 rows x 64 cols. A-tile and W staged in LDS.
// ---------------------------------------------------------------------------
template <int K, bool RELU>
__global__ __launch_bounds__(128) void gemm_wmma(const float* __restrict__ A,
                                                 const float* __restrict__ W,
                                                 const float* __restrict__ bias,
                                                 float* __restrict__ out) {
  __shared__ float sA[16 * K];
  __shared__ float sW[K * 64];
  const int tid = threadIdx.x;
  const int row0 = blockIdx.x * 16;
  for (int i = tid; i < 16 * K; i += 128) sA[i] = A[(size_t)row0 * K + i];
  for (int i = tid; i < K * 64; i += 128) sW[i] = W[i];
  __syncthreads();

  const int wave = tid >> 5;
  const int lane = tid & 31;
  const int col  = (wave << 4) + (lane & 15);  // output column 0..63
  const int hi   = lane >> 4;                  // half-wave select
  const int ar   = lane & 15;                  // A row within tile

  v8f acc = {0.f, 0.f, 0.f, 0.f, 0.f, 0.f, 0.f, 0.f};
#pragma unroll
  for (int k = 0; k < K; k += 4) {
    const int kk = k + (hi << 1);  // lanes 0-15 -> K=k,k+1 ; lanes 16-31 -> K=k+2,k+3
    v2f a, b;
    a.x = sA[ar * K + kk];
    a.y = sA[ar * K + kk + 1];
    b.x = sW[kk * 64 + col];
    b.y = sW[(kk + 1) * 64 + col];
    acc = __builtin_amdgcn_wmma_f32_16x16x4_f32(false, a, false, b, (short)0,
                                                acc, false, false);
  }

  const float bv = bias[col];
  const int rbase = row0 + (hi << 3);  // C/D layout: lanes>=16 hold rows M+8
#pragma unroll
  for (int r = 0; r < 8; ++r) {
    float v = acc[r] + bv;
    if (RELU) v = v > 0.f ? v : 0.f;
    out[(size_t)(rbase + r) * 64 + col] = v;
  }
}

// ---------------------------------------------------------------------------
// CSR construction (dst is invariant across layers -> build once per launch)
// ---------------------------------------------------------------------------
__global__ void zero_ints(int* __restrict__ p, int n) {
  int i = blockIdx.x * 256 + threadIdx.x;
  if (i < n) p[i] = 0;
}

__global__ void count_deg(const int* __restrict__ dst, int* __restrict__ deg, int e) {
  int i = blockIdx.x * 256 + threadIdx.x;
  if (i < e) atomicAdd(&deg[dst[i]], 1);
}

__global__ void scan_exclusive(const int* __restrict__ deg, int* __restrict__ row_ptr, int n) {
  __shared__ int s[1024];
  __shared__ int carry;
  const int tid = threadIdx.x;
  if (tid == 0) carry = 0;
  __syncthreads();
  const int rounds = (n + 1023) >> 10;
  for (int r = 0; r < rounds; ++r) {
    const int i = (r << 10) + tid;
    const int v = (i < n) ? deg[i] : 0;
    s[tid] = v;
    __syncthreads();
    for (int off = 1; off < 1024; off <<= 1) {
      const int t = (tid >= off) ? s[tid - off] : 0;
      __syncthreads();
      s[tid] += t;
      __syncthreads();
    }
    if (i < n) row_ptr[i] = carry + s[tid] - v;  // exclusive
    __syncthreads();
    if (tid == 1023) carry += s[1023];
    __syncthreads();
  }
  if (tid == 0) row_ptr[n] = carry;
}

__global__ void init_cur(const int* __restrict__ row_ptr, int* __restrict__ cur, int n) {
  int i = blockIdx.x * 256 + threadIdx.x;
  if (i < n) cur[i] = row_ptr[i];
}

__global__ void scatter_edges(const int* __restrict__ src, const int* __restrict__ dst,
                              int* __restrict__ cur, int* __restrict__ col_idx, int e) {
  int i = blockIdx.x * 256 + threadIdx.x;
  if (i < e) {
    const int p = atomicAdd(&cur[dst[i]], 1);
    col_idx[p] = src[i];
  }
}

// Canonicalize neighbor order (atomic scatter order is nondeterministic).
__global__ void sort_neighbors(int* __restrict__ col_idx, const int* __restrict__ row_ptr, int n) {
  int i = blockIdx.x * 256 + threadIdx.x;
  if (i >= n) return;
  const int s = row_ptr[i], e = row_ptr[i + 1];
  for (int a = s + 1; a < e; ++a) {
    const int key = col_idx[a];
    int b = a - 1;
    while (b >= s && col_idx[b] > key) { col_idx[b + 1] = col_idx[b]; --b; }
    col_idx[b + 1] = key;
  }
}

// ---------------------------------------------------------------------------
// GATv2 edge softmax + aggregation, one wave per node, online softmax.
// Lane owns channels 2*lane, 2*lane+1; head h lives in lanes 8h..8h+7.
// Self loop handled as the final iteration (matches reference edge order).
// ---------------------------------------------------------------------------
__global__ __launch_bounds__(256) void gat_agg(const float* __restrict__ xl,
                                               const float* __restrict__ xr,
                                               const float* __restrict__ att,
                                               const float* __restrict__ bg,
                                               const int* __restrict__ row_ptr,
                                               const int* __restrict__ col_idx,
                                               float* __restrict__ out, int n) {
  const int lane = threadIdx.x & 31;
  const int node = blockIdx.x * 8 + (threadIdx.x >> 5);
  if (node >= n) return;
  const int d0 = lane << 1, d1 = d0 + 1;
  const float xr0 = xr[(size_t)node * D + d0];
  const float xr1 = xr[(size_t)node * D + d1];
  const float a0 = att[d0], a1 = att[d1];
  float m = -__builtin_inff(), s = 0.f, acc0 = 0.f, acc1 = 0.f;
  const int beg = row_ptr[node], end = row_ptr[node + 1];
  for (int j = beg; j <= end; ++j) {
    const int src = (j < end) ? col_idx[j] : node;  // j==end -> self loop
    const float l0 = xl[(size_t)src * D + d0];
    const float l1 = xl[(size_t)src * D + d1];
    float t0 = l0 + xr0; t0 = t0 > 0.f ? t0 : NEG_SLOPE * t0;
    float t1 = l1 + xr1; t1 = t1 > 0.f ? t1 : NEG_SLOPE * t1;
    float e = t0 * a0 + t1 * a1;
    e += __shfl_xor(e, 1);
    e += __shfl_xor(e, 2);
    e += __shfl_xor(e, 4);  // per-head score broadcast to its 8 lanes
    const float mn = fmaxf(m, e);
    const float sc = __expf(m - mn);  // 0 on the first iteration (m = -inf)
    const float w  = __expf(e - mn);
    s    = s * sc + w;
    acc0 = acc0 * sc + w * l0;
    acc1 = acc1 * sc + w * l1;
    m = mn;
  }
  const float inv = 1.f / s;  // degree >= 1 (self loop), s > 0
  out[(size_t)node * D + d0] = acc0 * inv + bg[d0];
  out[(size_t)node * D + d1] = acc1 * inv + bg[d1];
}

// h = LN(tmp + h) * g + b, optional ReLU. One wave per node.
__global__ __launch_bounds__(256) void ln_residual(const float* __restrict__ tmp,
                                                   float* __restrict__ h,
                                                   const float* __restrict__ g,
                                                   const float* __restrict__ b,
                                                   int n, int do_relu) {
  const int lane = threadIdx.x & 31;
  const int node = blockIdx.x * 8 + (threadIdx.x >> 5);
  if (node >= n) return;
  const int d0 = lane << 1, d1 = d0 + 1;
  const float v0 = tmp[(size_t)node * D + d0] + h[(size_t)node * D + d0];
  const float v1 = tmp[(size_t)node * D + d1] + h[(size_t)node * D + d1];
  float sum = v0 + v1;
  for (int o = 1; o < 32; o <<= 1) sum += __shfl_xor(sum, o);
  const float mean = sum * (1.f / D);
  const float dv0 = v0 - mean, dv1 = v1 - mean;
  float vs = dv0 * dv0 + dv1 * dv1;
  for (int o = 1; o < 32; o <<= 1) vs += __shfl_xor(vs, o);
  const float rs = rsqrtf(vs * (1.f / D) + LN_EPS);
  float o0 = dv0 * rs * g[d0] + b[d0];
  float o1 = dv1 * rs * g[d1] + b[d1];
  if (do_relu) { o0 = fmaxf(o0, 0.f); o1 = fmaxf(o1, 0.f); }
  h[(size_t)node * D + d0] = o0;
  h[(size_t)node * D + d1] = o1;
}

// gate[i] = relu(h_i @ w_g1 + b_g1) @ w_g2 + b_g2 ; one wave per node,
// lane = one of the 32 hidden units.
__global__ __launch_bounds__(256) void gate_kernel(const float* __restrict__ h,
                                                   const float* __restrict__ w1,
                                                   const float* __restrict__ b1,
                                                   const float* __restrict__ w2,
                                                   const float* __restrict__ b2,
                                                   float* __restrict__ gate, int n) {
  const int lane = threadIdx.x & 31;
  const int node = blockIdx.x * 8 + (threadIdx.x >> 5);
  if (node >= n) return;
  const float* hrow = h + (size_t)node * D;
  float acc = b1[lane];
  for (int d = 0; d < D; ++d) acc += hrow[d] * w1[d * 32 + lane];
  acc = fmaxf(acc, 0.f);
  float t = acc * w2[lane];
  for (int o = 1; o < 32; o <<= 1) t += __shfl_xor(t, o);
  if (lane == 0) gate[node] = t + b2[0];
}

__device__ __forceinline__ int lower_bound_batch(const int* batch, int n, int g) {
  int lo = 0, hi = n;
  while (lo < hi) {
    const int mid = (lo + hi) >> 1;
    if (batch[mid] < g) lo = mid + 1; else hi = mid;
  }
  return lo;
}

// Per-graph softmax stats (batch is sorted -> binary search segment bounds).
__global__ __launch_bounds__(256) void graph_stats(const float* __restrict__ gate,
                                                   const int* __restrict__ batch,
                                                   float* __restrict__ gmax,
                                                   float* __restrict__ gs, int n) {
  __shared__ float red[256];
  __shared__ int sbeg, send;
  const int g = blockIdx.x;
  const int tid = threadIdx.x;
  if (tid == 0) { sbeg = lower_bound_batch(batch, n, g); send = lower_bound_batch(batch, n, g + 1); }
  __syncthreads();
  const int beg = sbeg, end = send;
  float mx = -__builtin_inff();
  for (int i = beg + tid; i < end; i += 256) mx = fmaxf(mx, gate[i]);
  red[tid] = mx; __syncthreads();
  for (int o = 128; o > 0; o >>= 1) { if (tid < o) red[tid] = fmaxf(red[tid], red[tid + o]); __syncthreads(); }
  const float m = red[0];
  __syncthreads();
  float sm = 0.f;
  for (int i = beg + tid; i < end; i += 256) sm += __expf(gate[i] - m);
  red[tid] = sm; __syncthreads();
  for (int o = 128; o > 0; o >>= 1) { if (tid < o) red[tid] += red[tid + o]; __syncthreads(); }
  if (tid == 0) { gmax[g] = m; gs[g] = red[0] > 0.f ? red[0] : 1.f; }
}

// emb[g] = sum_i softmax(gate)_i * h_i ; block per graph, 64 dims x 4 slices.
__global__ __launch_bounds__(256) void pool_kernel(const float* __restrict__ h,
                                                   const float* __restrict__ gate,
                                                   const int* __restrict__ batch,
                                                   const float* __restrict__ gmax,
                                                   const float* __restrict__ gs,
                                                   float* __restrict__ emb, int n) {
  __shared__ float red[256];
  __shared__ int sbeg, send;
  const int g = blockIdx.x;
  const int tid = threadIdx.x;
  if (tid == 0) { sbeg = lower_bound_batch(batch, n, g); send = lower_bound_batch(batch, n, g + 1); }
  __syncthreads();
  const int beg = sbeg, end = send;
  const int d = tid & 63, sl = tid >> 6;
  const float m = gmax[g], inv = 1.f / gs[g];
  float acc = 0.f;
  for (int i = beg + sl; i < end; i += 4)
    acc += __expf(gate[i] - m) * inv * h[(size_t)i * D + d];
  red[tid] = acc; __syncthreads();
  if (tid < 128) red[tid] += red[tid + 128];
  __syncthreads();
  if (tid < 64) { red[tid] += red[tid + 64]; emb[g * D + tid] = red[tid]; }
}

// ---------------------------------------------------------------------------
extern "C" void kernel_launch(void* const* d_in, const int* in_sizes, int n_in,
                              void* d_out, int out_size, void* d_ws, size_t ws_size,
                              hipStream_t stream) {
  const float* x     = (const float*)d_in[0];
  const int*   ei    = (const int*)d_in[1];
  const int*   batch = (const int*)d_in[2];
  const float* w_in  = (const float*)d_in[3];
  const float* b_in  = (const float*)d_in[4];
  const float* w_l   = (const float*)d_in[5];
  const float* b_l   = (const float*)d_in[6];
  const float* w_r   = (const float*)d_in[7];
  const float* b_r   = (const float*)d_in[8];
  const float* att   = (const float*)d_in[9];
  const float* b_gat = (const float*)d_in[10];
  const float* ln_g  = (const float*)d_in[11];
  const float* ln_b  = (const float*)d_in[12];
  const float* w_g1  = (const float*)d_in[13];
  const float* b_g1  = (const float*)d_in[14];
  const float* w_g2  = (const float*)d_in[15];
  const float* b_g2  = (const float*)d_in[16];
  const float* w_out = (const float*)d_in[17];
  const float* b_out = (const float*)d_in[18];

  const int N = in_sizes[2];       // 50000 (multiple of 16)
  const int E = in_sizes[1] / 2;   // 1600000
  const int G = out_size / D;      // 64
  const int* esrc = ei;
  const int* edst = ei + E;

  // Workspace carve-up (floats first, then ints).
  float* fws = (float*)d_ws;
  size_t off = 0;
  float* h    = fws + off; off += (size_t)N * D;
  float* tmp  = fws + off; off += (size_t)N * D;
  float* xl   = fws + off; off += (size_t)N * D;
  float* xr   = fws + off; off += (size_t)N * D;
  float* gate = fws + off; off += (size_t)N;
  float* gmax = fws + off; off += (size_t)G;
  float* gs   = fws + off; off += (size_t)G;
  float* emb  = fws + off; off += (size_t)G * D;
  int* iws     = (int*)(fws + off);
  int* deg     = iws; iws += N;
  int* row_ptr = iws; iws += (N + 1);
  int* cur     = iws; iws += N;
  int* col_idx = iws; iws += E;

  const int nb_n = (N + 255) / 256;
  const int nb_e = (E + 255) / 256;
  const int nb_w = N / 8;  // wave-per-node kernels (N % 8 == 0)

  // 1) Build CSR by destination (layer-invariant).
  zero_ints<<<nb_n, 256, 0, stream>>>(deg, N);
  count_deg<<<nb_e, 256, 0, stream>>>(edst, deg, E);
  scan_exclusive<<<1, 1024, 0, stream>>>(deg, row_ptr, N);
  init_cur<<<nb_n, 256, 0, stream>>>(row_ptr, cur, N);
  scatter_edges<<<nb_e, 256, 0, stream>>>(esrc, edst, cur, col_idx, E);
  sort_neighbors<<<nb_n, 256, 0, stream>>>(col_idx, row_ptr, N);

  // 2) Input projection: h = relu(x @ w_in + b_in), K = 24.
  gemm_wmma<24, true><<<N / 16, 128, 0, stream>>>(x, w_in, b_in, h);

  // 3) Three GATv2 layers.
  for (int l = 0; l < 3; ++l) {
    gemm_wmma<64, false><<<N / 16, 128, 0, stream>>>(h, w_l + (size_t)l * D * D,
                                                     b_l + l * D, xl);
    gemm_wmma<64, false><<<N / 16, 128, 0, stream>>>(h, w_r + (size_t)l * D * D,
                                                     b_r + l * D, xr);
    gat_agg<<<nb_w, 256, 0, stream>>>(xl, xr, att + l * D, b_gat + l * D,
                                      row_ptr, col_idx, tmp, N);
    ln_residual<<<nb_w, 256, 0, stream>>>(tmp, h, ln_g + l * D, ln_b + l * D, N,
                                          (l < 2) ? 1 : 0);
  }

  // 4) Global attention pooling.
  gate_kernel<<<nb_w, 256, 0, stream>>>(h, w_g1, b_g1, w_g2, b_g2, gate, N);
  graph_stats<<<G, 256, 0, stream>>>(gate, batch, gmax, gs, N);
  pool_kernel<<<G, 256, 0, stream>>>(h, gate, batch, gmax, gs, emb, N);

  // 5) Readout: out = relu(emb @ w_out + b_out), G x 64 rows (G % 16 == 0).
  gemm_wmma<64, true><<<G / 16, 128, 0, stream>>>(emb, w_out, b_out, (float*)d_out);
}